// EmotionConstraintLoss_55954833933114
// MI455X (gfx1250) — compile-verified
//
#include <hip/hip_runtime.h>
#include <math.h>

typedef __attribute__((ext_vector_type(2))) float v2f;
typedef __attribute__((ext_vector_type(8))) float v8f;

#define B_N 8192
#define D_N 1024
#define C_N 8

// Workspace layout (floats):
//   [0]            ce_sum
//   [1]            center_sum
//   [2..10)        class counts (as float; integer counts <= 8192 are exact)
//   [16..16+16*1024)   G: class sums of normalized features, padded to 16 rows
//   [16400..16400+8192) inv_norm per row
#define WS_CE     0
#define WS_CENTER 1
#define WS_CNT    2
#define WS_G      16
#define WS_INV    (16 + 16 * D_N)
#define WS_ZERO_N (16 + 16 * D_N)   // region that must start at 0

__global__ void k_zero(float* __restrict__ ws) {
    int i = blockIdx.x * 256 + threadIdx.x;
    if (i < WS_ZERO_N) ws[i] = 0.0f;
}

// One wave32 per row: squared-norm reduction -> inv_norm; lane 0 also does the
// 8-class CE term and the class-count bump.
__global__ void k_norm_ce(const float* __restrict__ feat,
                          const int* __restrict__ labels,
                          const float* __restrict__ logits,
                          float* __restrict__ ws) {
    int wave = threadIdx.x >> 5;
    int lane = threadIdx.x & 31;
    int i = blockIdx.x * 8 + wave;

    const float4* row = (const float4*)(feat + (size_t)i * D_N);
    float s = 0.0f;
#pragma unroll
    for (int t = 0; t < 8; ++t) {              // 256 float4 per row / 32 lanes
        float4 v = row[lane + 32 * t];
        s += v.x * v.x + v.y * v.y + v.z * v.z + v.w * v.w;
    }
#pragma unroll
    for (int off = 16; off >= 1; off >>= 1)
        s += __shfl_xor(s, off, 32);

    if (lane == 0) {
        ws[WS_INV + i] = 1.0f / fmaxf(sqrtf(s), 1e-8f);

        const float* lg = logits + i * C_N;
        float m = lg[0];
#pragma unroll
        for (int c = 1; c < C_N; ++c) m = fmaxf(m, lg[c]);
        float se = 0.0f;
#pragma unroll
        for (int c = 0; c < C_N; ++c) se += expf(lg[c] - m);
        int lab = labels[i];
        float ce = (m + logf(se)) - lg[lab];
        atomicAdd(&ws[WS_CE], ce);
        atomicAdd(&ws[WS_CNT + lab], 1.0f);
    }
}

// Column-parallel pass: thread owns one dim d, walks 256 rows.
// Accumulates per-class sums of fn = f * inv_norm, plus the center loss.
__global__ void k_gsum(const float* __restrict__ feat,
                       const int* __restrict__ labels,
                       const float* __restrict__ centers,
                       float* __restrict__ ws) {
    int d  = blockIdx.x * 256 + threadIdx.x;   // 0..1023
    int r0 = blockIdx.y * 256;
    const float* inv = ws + WS_INV;

    float acc[C_N];
#pragma unroll
    for (int c = 0; c < C_N; ++c) acc[c] = 0.0f;
    float csum = 0.0f;

    for (int r = 0; r < 256; ++r) {
        int i = r0 + r;
        int lab = labels[i];                   // scalar broadcast
        float w = inv[i];                      // scalar broadcast
        float v = feat[(size_t)i * D_N + d];   // coalesced across lanes
        float fn = v * w;
#pragma unroll
        for (int c = 0; c < C_N; ++c)
            acc[c] += (lab == c) ? fn : 0.0f;
        float diff = v - centers[lab * D_N + d];
        csum += diff * diff;
    }

#pragma unroll
    for (int c = 0; c < C_N; ++c)
        atomicAdd(&ws[WS_G + c * D_N + d], acc[c]);

    __shared__ float red[256];
    red[threadIdx.x] = csum;
    __syncthreads();
    for (int s2 = 128; s2 >= 1; s2 >>= 1) {
        if (threadIdx.x < (unsigned)s2) red[threadIdx.x] += red[threadIdx.x + s2];
        __syncthreads();
    }
    if (threadIdx.x == 0) atomicAdd(&ws[WS_CENTER], red[0]);
}

// Single wave32: M = G * G^T via V_WMMA_F32_16X16X4_F32 (exact f32), then the
// S-weighted contraction and final scalar assembly.
//
// ISA layouts (32-bit, 16x16x4):
//   A 16x4:  lane L<16 -> {A[L][k0], A[L][k0+1]}, lane L+16 -> {A[L][k0+2], A[L][k0+3]}
//   B 4x16:  v0: K=k0 (lanes 0-15) / K=k0+2 (lanes 16-31); v1: K=k0+1 / K=k0+3
// For B = G^T both operand tiles hold identical per-lane data -> one v2f load.
__global__ void k_final(const float* __restrict__ S,
                        float* __restrict__ ws,
                        float* __restrict__ out) {
    const float* G = ws + WS_G;                // 16 x 1024, rows 8..15 are zero
    int lane = threadIdx.x;
    int r    = lane & 15;
    int koff = (lane >> 4) << 1;               // 0 or 2
    const float* Grow = G + r * D_N + koff;

    v8f acc = {};
    for (int k0 = 0; k0 < D_N; k0 += 4) {
        v2f ab;
        ab.x = Grow[k0];
        ab.y = Grow[k0 + 1];
        acc = __builtin_amdgcn_wmma_f32_16x16x4_f32(
            /*neg_a=*/false, ab, /*neg_b=*/false, ab,
            /*c_mod=*/(short)0, acc, /*reuse_a=*/false, /*reuse_b=*/false);
    }

    // D layout: VGPR v, lanes 0-15 -> M=v, N=lane; lanes 16-31 -> M=v+8, N=lane-16
    int m0 = (lane < 16) ? 0 : 8;
    int n  = lane & 15;
    float p = 0.0f;
#pragma unroll
    for (int v = 0; v < 8; ++v) {
        int m = m0 + v;
        float sv = S[(m & 7) * C_N + (n & 7)]; // always in-bounds
        if (m >= C_N || n >= C_N) sv = 0.0f;   // padded rows/cols contribute 0
        p += sv * acc[v];
    }
#pragma unroll
    for (int off = 16; off >= 1; off >>= 1)
        p += __shfl_xor(p, off, 32);

    if (lane == 0) {
        float cnt[C_N];
        for (int c = 0; c < C_N; ++c) cnt[c] = ws[WS_CNT + c];
        float W1 = 0.0f;
        for (int a = 0; a < C_N; ++a)
            for (int b = 0; b < C_N; ++b)
                W1 += cnt[a] * cnt[b] * S[a * C_N + b];

        float Bf     = (float)B_N;
        float ce     = ws[WS_CE] / Bf;
        float center = ws[WS_CENTER] / (2.0f * Bf);
        float soft   = (W1 - p) / (Bf * (Bf - 1.0f));
        out[0] = ce + center + soft;
    }
}

extern "C" void kernel_launch(void* const* d_in, const int* in_sizes, int n_in,
                              void* d_out, int out_size, void* d_ws, size_t ws_size,
                              hipStream_t stream) {
    const float* feat    = (const float*)d_in[0];   // [8192,1024]
    const int*   labels  = (const int*)d_in[1];     // [8192]
    const float* logits  = (const float*)d_in[2];   // [8192,8]
    const float* S       = (const float*)d_in[3];   // [8,8]
    const float* centers = (const float*)d_in[4];   // [8,1024]
    float* out = (float*)d_out;
    float* ws  = (float*)d_ws;

    hipLaunchKernelGGL(k_zero, dim3((WS_ZERO_N + 255) / 256), dim3(256), 0, stream, ws);
    hipLaunchKernelGGL(k_norm_ce, dim3(B_N / 8), dim3(256), 0, stream,
                       feat, labels, logits, ws);
    hipLaunchKernelGGL(k_gsum, dim3(D_N / 256, B_N / 256), dim3(256), 0, stream,
                       feat, labels, centers, ws);
    hipLaunchKernelGGL(k_final, dim3(1), dim3(32), 0, stream, S, ws, out);
}